// QuantumImageEmbedder_21844203668210
// MI455X (gfx1250) — compile-verified
//
#include <hip/hip_runtime.h>
#include <hip/hip_bf16.h>
#include <math.h>

// ---------------- complex helpers ----------------
struct cpx { float re, im; };
__device__ inline cpx cmk(float r, float i){ cpx c; c.re=r; c.im=i; return c; }
__device__ inline cpx cadd(cpx a, cpx b){ return cmk(a.re+b.re, a.im+b.im); }
__device__ inline cpx csub(cpx a, cpx b){ return cmk(a.re-b.re, a.im-b.im); }
__device__ inline cpx cmul(cpx a, cpx b){ return cmk(a.re*b.re - a.im*b.im, a.re*b.im + a.im*b.re); }
__device__ inline cpx ccj (cpx a){ return cmk(a.re, -a.im); }
__device__ inline cpx cscl(cpx a, float s){ return cmk(a.re*s, a.im*s); }

struct mat2 { cpx m[2][2]; };

__device__ inline mat2 mrx(float t){
  float c=cosf(0.5f*t), s=sinf(0.5f*t);
  mat2 M; M.m[0][0]=cmk(c,0); M.m[0][1]=cmk(0,-s); M.m[1][0]=cmk(0,-s); M.m[1][1]=cmk(c,0); return M;
}
__device__ inline mat2 mry(float t){
  float c=cosf(0.5f*t), s=sinf(0.5f*t);
  mat2 M; M.m[0][0]=cmk(c,0); M.m[0][1]=cmk(-s,0); M.m[1][0]=cmk(s,0); M.m[1][1]=cmk(c,0); return M;
}
__device__ inline mat2 mrz(float t){
  float c=cosf(0.5f*t), s=sinf(0.5f*t);
  mat2 M; M.m[0][0]=cmk(c,-s); M.m[0][1]=cmk(0,0); M.m[1][0]=cmk(0,0); M.m[1][1]=cmk(c,s); return M;
}
__device__ inline mat2 mmul(const mat2&A, const mat2&B){
  mat2 C;
  #pragma unroll
  for (int i=0;i<2;++i)
    #pragma unroll
    for (int j=0;j<2;++j)
      C.m[i][j] = cadd(cmul(A.m[i][0],B.m[0][j]), cmul(A.m[i][1],B.m[1][j]));
  return C;
}
__device__ inline void mvec(const mat2&A, cpx&v0, cpx&v1){
  cpx n0 = cadd(cmul(A.m[0][0],v0), cmul(A.m[0][1],v1));
  cpx n1 = cadd(cmul(A.m[1][0],v0), cmul(A.m[1][1],v1));
  v0=n0; v1=n1;
}

// ------------- expectation via GF(2) contraction -------------
// psi = P_L |v0 ... v19>, L from the CNOT ring. Product observable on wires
// W (subset of {0..3}), O[w] is the 2x2 observable applied on wire w.
// E = sum_{d in Null} sum_{S subset W} coeff(d,S) * prod_j g_j(d_j, r_j)
__device__ float expect_dev(const cpx (*V)[2], const cpx (*O)[2][2], unsigned wmask){
  // row masks of L: bit_w(Lx) = <m[w], x>
  unsigned m[20];
  #pragma unroll
  for (int j=0;j<20;++j) m[j]=1u<<j;
  #pragma unroll
  for (int i=0;i<20;++i) m[(i+1)%20] ^= m[i];

  // constraints: <m[w], d> = 0 for all identity wires
  unsigned rows[20]; int nr=0;
  for (int w=0;w<20;++w) if (w>=4 || !((wmask>>w)&1)) rows[nr++]=m[w];

  // GF(2) reduced row echelon -> nullspace basis
  int pivcol[20]; int rr=0;
  for (int col=0; col<20 && rr<nr; ++col){
    int sel=-1;
    for (int i=rr;i<nr;++i) if ((rows[i]>>col)&1){ sel=i; break; }
    if (sel<0) continue;
    unsigned t=rows[rr]; rows[rr]=rows[sel]; rows[sel]=t;
    for (int i=0;i<nr;++i) if (i!=rr && ((rows[i]>>col)&1)) rows[i]^=rows[rr];
    pivcol[rr]=col; ++rr;
  }
  bool ispiv[20];
  for (int c=0;c<20;++c) ispiv[c]=false;
  for (int i=0;i<rr;++i) ispiv[pivcol[i]]=true;
  unsigned basis[8]; int nb=0;
  for (int col=0; col<20; ++col){
    if (ispiv[col]) continue;
    unsigned vv = 1u<<col;
    for (int i=0;i<rr;++i) if ((rows[i]>>col)&1) vv |= 1u<<pivcol[i];
    if (nb<8) basis[nb++]=vv;
  }

  int wl[4]; int nw=0;
  for (int w=0;w<4;++w) if ((wmask>>w)&1) wl[nw++]=w;

  cpx E = cmk(0,0);
  for (int di=0; di<(1<<nb); ++di){
    unsigned d=0;
    for (int k=0;k<nb;++k) if ((di>>k)&1) d ^= basis[k];
    // per-wire decomposition O[b][b^c] = alpha + beta*(-1)^b
    cpx al[4], be[4];
    for (int t=0;t<nw;++t){
      int w = wl[t];
      int cw = __builtin_popcount(m[w]&d)&1;
      cpx f0 = O[w][0][cw];
      cpx f1 = O[w][1][1^cw];
      al[t] = cscl(cadd(f0,f1),0.5f);
      be[t] = cscl(csub(f0,f1),0.5f);
    }
    for (int S=0; S<(1<<nw); ++S){
      cpx coeff = cmk(1,0); unsigned r=0;
      for (int t=0;t<nw;++t){
        if ((S>>t)&1){ coeff = cmul(coeff, be[t]); r ^= m[wl[t]]; }
        else         { coeff = cmul(coeff, al[t]); }
      }
      cpx prod = cmk(1,0);
      #pragma unroll
      for (int j=0;j<20;++j){
        int dj=(d>>j)&1, rj=(r>>j)&1;
        cpx g = cmul(ccj(V[j][0]), V[j][dj]);
        cpx h = cmul(ccj(V[j][1]), V[j][1^dj]);
        g = rj ? csub(g,h) : cadd(g,h);
        prod = cmul(prod,g);
      }
      E = cadd(E, cmul(coeff, prod));
    }
  }
  return E.re;
}

// ------------- kernel 1: per-image quantum expectations -------------
__global__ __launch_bounds__(96)
void qexp_kernel(const float* __restrict__ x, const float* __restrict__ posw,
                 const float* __restrict__ l1w, const float* __restrict__ l2w,
                 float* __restrict__ qbuf){
  __shared__ cpx sV[4][20][2];     // per-block per-qubit states after layer 1 locals
  __shared__ cpx sO[16][2][2];     // pulled-back layer-2 observables
  __shared__ float sE[7];
  int img = blockIdx.x;
  int tid = threadIdx.x;

  if (tid < 80){
    int b = tid/20, i = tid%20;
    int p = 4*b + i/5, j = i%5;
    cpx v0 = cmk(1,0), v1 = cmk(0,0);
    if (j < 4){
      const float* a = x + (((img*16 + p)*4) + j)*3;
      mat2 Mx = mrx(a[0]); mvec(Mx, v0, v1);
      mat2 My = mry(a[1]); mvec(My, v0, v1);
      mat2 Mz = mrz(a[2]); mvec(Mz, v0, v1);
    } else {
      mat2 Mx = mrx(posw[p*2+0]); mvec(Mx, v0, v1);
      mat2 My = mry(posw[p*2+1]); mvec(My, v0, v1);
    }
    mat2 Lx = mrx(l1w[i*3+0]); mvec(Lx, v0, v1);
    mat2 Ly = mry(l1w[i*3+1]); mvec(Ly, v0, v1);
    mat2 Lz = mrz(l1w[i*3+2]); mvec(Lz, v0, v1);
    sV[b][i][0]=v0; sV[b][i][1]=v1;
  } else {
    int q = tid - 80;              // 0..15
    mat2 U = mmul(mrz(l2w[q*3+2]), mmul(mry(l2w[q*3+1]), mrx(l2w[q*3+0])));
    // O = U^dag Z U  ->  O[i][j] = conj(U[0][i])U[0][j] - conj(U[1][i])U[1][j]
    #pragma unroll
    for (int i=0;i<2;++i)
      #pragma unroll
      for (int j=0;j<2;++j)
        sO[q][i][j] = csub(cmul(ccj(U.m[0][i]), U.m[0][j]),
                           cmul(ccj(U.m[1][i]), U.m[1][j]));
  }
  __syncthreads();

  if (tid < 7){
    unsigned wmask; int blk;
    if (tid < 4)      { wmask = 0xFu; blk = tid; }  // Efull[0..3]
    else if (tid == 4){ wmask = 0xEu; blk = 0;  }   // wires 1,2,3 on block 0
    else if (tid == 5){ wmask = 0x3u; blk = 0;  }   // wires 0,1
    else              { wmask = 0x7u; blk = 0;  }   // wires 0,1,2
    sE[tid] = expect_dev(sV[blk], sO + 4*blk, wmask);
  }
  __syncthreads();

  if (tid == 0){
    qbuf[img*4+0] = sE[4]*sE[1]*sE[2]*sE[3];  // e0
    qbuf[img*4+1] = sE[5];                    // e1
    qbuf[img*4+2] = sE[6];                    // e2
    qbuf[img*4+3] = sE[0];                    // e3 = Efull[0]
  }
}

// ------------- kernel 2: WMMA head matmul + row normalize -------------
typedef float v2f __attribute__((ext_vector_type(2)));
typedef float v8f __attribute__((ext_vector_type(8)));

__global__ __launch_bounds__(32)
void head_kernel(const float* __restrict__ qbuf, const float* __restrict__ hw,
                 const float* __restrict__ hb, float* __restrict__ out){
  __shared__ float ys[8][512];
  __shared__ float nrm2[8];
  int l = threadIdx.x;                 // 0..31, one wave
  int row   = (l < 16) ? l : (l - 16);
  int kbase = (l < 16) ? 0 : 2;

  // A fragment: 16x4, rows 0..7 = q (images), rows 8..15 zero pad
  v2f a;
  a.x = (row < 8) ? qbuf[row*4 + kbase + 0] : 0.0f;
  a.y = (row < 8) ? qbuf[row*4 + kbase + 1] : 0.0f;

  for (int t=0; t<32; ++t){
    int n0 = t*16;
    // B fragment: B[k][n] = head_w[n0+n][k]
    int n = n0 + row;
    v2f b;
    b.x = hw[n*4 + kbase + 0];
    b.y = hw[n*4 + kbase + 1];
    v8f c = {};
    c = __builtin_amdgcn_wmma_f32_16x16x4_f32(
        /*neg_a=*/false, a, /*neg_b=*/false, b,
        /*c_mod=*/(short)0, c, /*reuse_a=*/false, /*reuse_b=*/false);
    if (l < 16){
      float bias = hb[n0 + l];
      #pragma unroll
      for (int r=0; r<8; ++r) ys[r][n0 + l] = c[r] + bias;  // lane<16 -> row r
    }
  }
  __syncthreads();

  #pragma unroll
  for (int r=0; r<8; ++r){
    float s = 0.0f;
    for (int n=l; n<512; n+=32){ float yv = ys[r][n]; s += yv*yv; }
    #pragma unroll
    for (int off=16; off>0; off>>=1) s += __shfl_down(s, off, 32);
    if (l == 0) nrm2[r] = s;
  }
  __syncthreads();

  #pragma unroll
  for (int r=0; r<8; ++r){
    float inv = 1.0f / fmaxf(sqrtf(nrm2[r]), 1e-12f);
    for (int n=l; n<512; n+=32) out[r*512 + n] = ys[r][n] * inv;
  }
}

extern "C" void kernel_launch(void* const* d_in, const int* in_sizes, int n_in,
                              void* d_out, int out_size, void* d_ws, size_t ws_size,
                              hipStream_t stream) {
  const float* x    = (const float*)d_in[0];  // [8,16,4,3]
  const float* posw = (const float*)d_in[1];  // [16,2]
  const float* l1w  = (const float*)d_in[2];  // [20,3]
  const float* l2w  = (const float*)d_in[3];  // [16,3]
  const float* hw   = (const float*)d_in[4];  // [512,4]
  const float* hb   = (const float*)d_in[5];  // [512]
  float* out  = (float*)d_out;                // [8,512]
  float* qbuf = (float*)d_ws;                 // [8,4] scratch

  qexp_kernel<<<8, 96, 0, stream>>>(x, posw, l1w, l2w, qbuf);
  head_kernel<<<1, 32, 0, stream>>>(qbuf, hw, hb, out);
}